// Cbp_layer_59897613910676
// MI455X (gfx1250) — compile-verified
//
#include <hip/hip_runtime.h>

typedef __attribute__((ext_vector_type(2))) float v2f;
typedef __attribute__((ext_vector_type(8))) float v8f;

#define BATCH 32
#define HW    196      // 14*14
#define CH    512
#define DD    8192

__global__ void cbp_zero_out(float* __restrict__ out, int n) {
  int i = blockIdx.x * blockDim.x + threadIdx.x;
  if (i < n) out[i] = 0.0f;
}

// One wave computes a 32x32 tile of G_b = X1_b^T * X2_b  (M=N=512, K=196)
// via V_WMMA_F32_16X16X4_F32, then scatters its tile directly into out[b, :]
// with global_atomic_add_f32 at bucket (h1[m]+h2[n]) & (D-1), signed s1[m]*s2[n].
__global__ __launch_bounds__(128)
void cbp_gemm_scatter(const float* __restrict__ x1,
                      const float* __restrict__ x2,
                      const float* __restrict__ s1,
                      const float* __restrict__ s2,
                      const int*   __restrict__ h1,
                      const int*   __restrict__ h2,
                      float* __restrict__ out) {
  const int lane   = threadIdx.x & 31;
  const int half   = lane >> 4;          // 0: lanes 0-15, 1: lanes 16-31
  const int l      = lane & 15;
  const int waveId = blockIdx.x * 4 + (threadIdx.x >> 5);

  // 16x16 = 256 wave-tiles per batch (32x32 elements each)
  const int b  = waveId >> 8;
  const int t  = waveId & 255;
  const int m0 = (t >> 4) << 5;          // c1 tile base
  const int n0 = (t & 15) << 5;          // c2 tile base

  const float* x1b = x1 + (size_t)b * HW * CH;
  const float* x2b = x2 + (size_t)b * HW * CH;

  v8f acc00 = {}, acc01 = {}, acc10 = {}, acc11 = {};

  // A-frag (16x4 f32): lane<16 holds K=k0+0 (v0), k0+1 (v1); lane>=16 holds K=k0+2, k0+3.
  // B-frag (4x16 f32): VGPR j holds K=kbase+j, column = lane&15.  kbase = 2*half.
  const int kb = 2 * half;
  for (int k0 = 0; k0 < HW; k0 += 4) {
    const float* r0 = x1b + (size_t)(k0 + kb) * CH;   // X1 row k
    const float* r1 = r0 + CH;                        // X1 row k+1
    v2f a0 = { r0[m0 + l],      r1[m0 + l]      };
    v2f a1 = { r0[m0 + 16 + l], r1[m0 + 16 + l] };
    const float* q0 = x2b + (size_t)(k0 + kb) * CH;   // X2 row k
    const float* q1 = q0 + CH;
    v2f b0 = { q0[n0 + l],      q1[n0 + l]      };
    v2f b1 = { q0[n0 + 16 + l], q1[n0 + 16 + l] };

    acc00 = __builtin_amdgcn_wmma_f32_16x16x4_f32(false, a0, false, b0, (short)0, acc00, false, false);
    acc01 = __builtin_amdgcn_wmma_f32_16x16x4_f32(false, a0, false, b1, (short)0, acc01, false, false);
    acc10 = __builtin_amdgcn_wmma_f32_16x16x4_f32(false, a1, false, b0, (short)0, acc10, false, false);
    acc11 = __builtin_amdgcn_wmma_f32_16x16x4_f32(false, a1, false, b1, (short)0, acc11, false, false);
  }

  // C/D layout (32-bit 16x16): VGPR r holds M = r + 8*half (+ tile base), N = lane&15 (+ tile base).
  float* outb = out + (size_t)b * DD;
  const int nA = n0 + l;
  const int nB = n0 + 16 + l;
  const int   hn0 = h2[nA];  const float sn0 = s2[nA];
  const int   hn1 = h2[nB];  const float sn1 = s2[nB];

  #pragma unroll
  for (int mi = 0; mi < 2; ++mi) {
    #pragma unroll
    for (int r = 0; r < 8; ++r) {
      const int   m  = m0 + mi * 16 + 8 * half + r;
      const int   hm = h1[m];
      const float sm = s1[m];
      const float g0 = (mi == 0) ? acc00[r] : acc10[r];
      const float g1 = (mi == 0) ? acc01[r] : acc11[r];
      atomicAdd(&outb[(hm + hn0) & (DD - 1)], g0 * sm * sn0);
      atomicAdd(&outb[(hm + hn1) & (DD - 1)], g1 * sm * sn1);
    }
  }
}

extern "C" void kernel_launch(void* const* d_in, const int* in_sizes, int n_in,
                              void* d_out, int out_size, void* d_ws, size_t ws_size,
                              hipStream_t stream) {
  const float* x1 = (const float*)d_in[0];   // bottom1 [32,14,14,512] f32
  const float* x2 = (const float*)d_in[1];   // bottom2 [32,14,14,512] f32
  const float* s1 = (const float*)d_in[2];   // rand_s_1 [512] f32 (+/-1)
  const float* s2 = (const float*)d_in[3];   // rand_s_2 [512] f32
  const int*   h1 = (const int*)d_in[4];     // rand_h_1 [512] i32 in [0,8192)
  const int*   h2 = (const int*)d_in[5];     // rand_h_2 [512] i32
  float* out = (float*)d_out;                // [32, 8192] f32

  // Harness poisons d_out; zero it before the atomic scatter.
  cbp_zero_out<<<(out_size + 255) / 256, 256, 0, stream>>>(out, out_size);

  // 32 batches * 16 * 16 tiles = 8192 waves; 4 waves (128 threads) per block.
  const int numWaves = BATCH * (CH / 32) * (CH / 32);
  cbp_gemm_scatter<<<numWaves / 4, 128, 0, stream>>>(x1, x2, s1, s2, h1, h2, out);
}